// BaseModel_27857157882299
// MI455X (gfx1250) — compile-verified
//
#include <hip/hip_runtime.h>

// GCN forward: h1 = relu(agg(X@W1)+b1); h2 = relu(agg(h1@W2)+b2);
// out = relu(h2@Wl1+bl1)@Wl2 + bl2
// Strategy:
//  - build dst-CSR once per call (int atomics only), reuse for both conv layers
//  - aggregation: one wave32 per node, one lane per channel (HID==32), gather-based,
//    unrolled x4 with independent accumulators for memory-level parallelism
//  - dense GEMMs + MLP head: V_WMMA_F32_16X16X4_F32 (fp32 precision preserved)

#define N_NODES 100000
#define N_EDGES 6400000
#define IN_CH   128
#define HID     32
#define SCAN_B  256
#define NBLK    ((N_NODES + SCAN_B - 1) / SCAN_B)   // 391

typedef __attribute__((ext_vector_type(2))) float v2f;
typedef __attribute__((ext_vector_type(8))) float v8f;

// ----------------------------- degree / CSR build -----------------------------

__global__ void k_zero_int(int* __restrict__ p, int n) {
  int i = blockIdx.x * blockDim.x + threadIdx.x;
  if (i < n) p[i] = 0;
}

__global__ void k_count(const long long* __restrict__ dst, int* __restrict__ cnt) {
  int e = blockIdx.x * blockDim.x + threadIdx.x;
  if (e < N_EDGES) atomicAdd(&cnt[(int)dst[e]], 1);
}

__global__ void k_dinv(const int* __restrict__ cnt, float* __restrict__ dinv) {
  int i = blockIdx.x * blockDim.x + threadIdx.x;
  if (i < N_NODES) dinv[i] = rsqrtf((float)cnt[i] + 1.0f);  // +1 = self loop
}

__global__ void k_block_sums(const int* __restrict__ cnt, int* __restrict__ bsum) {
  __shared__ int sh[SCAN_B];
  int tid = threadIdx.x;
  int i = blockIdx.x * SCAN_B + tid;
  sh[tid] = (i < N_NODES) ? cnt[i] : 0;
  __syncthreads();
  for (int s = SCAN_B / 2; s > 0; s >>= 1) {
    if (tid < s) sh[tid] += sh[tid + s];
    __syncthreads();
  }
  if (tid == 0) bsum[blockIdx.x] = sh[0];
}

__global__ void k_scan_sums(int* __restrict__ bsum) {
  if (threadIdx.x == 0 && blockIdx.x == 0) {
    int run = 0;
    for (int b = 0; b < NBLK; ++b) { int t = bsum[b]; bsum[b] = run; run += t; }
  }
}

__global__ void k_scan_final(const int* __restrict__ cnt, const int* __restrict__ bsum,
                             int* __restrict__ off, int* __restrict__ cur) {
  __shared__ int sh[SCAN_B];
  int tid = threadIdx.x;
  int i = blockIdx.x * SCAN_B + tid;
  int v = (i < N_NODES) ? cnt[i] : 0;
  sh[tid] = v;
  __syncthreads();
  for (int d = 1; d < SCAN_B; d <<= 1) {
    int t = (tid >= d) ? sh[tid - d] : 0;
    __syncthreads();
    sh[tid] += t;
    __syncthreads();
  }
  if (i < N_NODES) {
    int excl = bsum[blockIdx.x] + sh[tid] - v;
    off[i] = excl;
    cur[i] = excl;
  }
  if (i == 0) off[N_NODES] = N_EDGES;
}

__global__ void k_csr_fill(const long long* __restrict__ src, const long long* __restrict__ dst,
                           int* __restrict__ cur, int* __restrict__ csr_src) {
  int e = blockIdx.x * blockDim.x + threadIdx.x;
  if (e < N_EDGES) {
    int d = (int)dst[e];
    int pos = atomicAdd(&cur[d], 1);
    csr_src[pos] = (int)src[e];
  }
}

// ----------------------------- dense GEMM via fp32 WMMA -----------------------------
// H[M x 32] = X[M x K] @ W[K x 32]; one wave per 16-row tile, both N-tiles.
// A frag (16x4 f32): lane = half*16 + m; a = { X[m][k+2*half], X[m][k+2*half+1] }
// B frag (4x16 f32): lane = half*16 + n; b = { W[k+2*half][n], W[k+2*half+1][n] }
// C/D (16x16 f32):   lanes 0-15 rows 0-7 (vgpr=row), lanes 16-31 rows 8-15

template <int K>
__global__ void __launch_bounds__(32) k_gemm_wmma(const float* __restrict__ X,
                                                  const float* __restrict__ W,
                                                  float* __restrict__ H) {
  const int m0   = blockIdx.x * 16;
  const int lane = threadIdx.x;
  const int half = lane >> 4;
  const int l    = lane & 15;
  v8f c0 = {};
  v8f c1 = {};
  const float* xrow = X + (size_t)(m0 + l) * K;
  for (int k = 0; k < K; k += 4) {
    const int ka = k + 2 * half;
    v2f a;  a.x  = xrow[ka];              a.y  = xrow[ka + 1];
    v2f b0; b0.x = W[ka * HID + l];       b0.y = W[(ka + 1) * HID + l];
    v2f b1; b1.x = W[ka * HID + 16 + l];  b1.y = W[(ka + 1) * HID + 16 + l];
    c0 = __builtin_amdgcn_wmma_f32_16x16x4_f32(false, a, false, b0, (short)0, c0, false, false);
    c1 = __builtin_amdgcn_wmma_f32_16x16x4_f32(false, a, false, b1, (short)0, c1, false, false);
  }
#pragma unroll
  for (int v = 0; v < 8; ++v) {
    const int r = half * 8 + v;
    H[(size_t)(m0 + r) * HID + l]      = c0[v];
    H[(size_t)(m0 + r) * HID + 16 + l] = c1[v];
  }
}

// ----------------------------- gather aggregation -----------------------------
// out[v] = relu( sum_{e:dst=v} h[src_e]*dinv[src_e]*dinv[v] + h[v]*dinv[v]^2 + b )
// one wave per node; lane == feature channel (HID == 32 == wave32).
// Unrolled x4 with independent accumulators: 4 csr loads + 4 dinv loads + 4
// coalesced 128B row loads in flight per trip (claused by the compiler).

__global__ void k_aggregate(const float* __restrict__ h, const float* __restrict__ dinv,
                            const int* __restrict__ off, const int* __restrict__ csr_src,
                            const float* __restrict__ bias, float* __restrict__ out) {
  const int node = (blockIdx.x * blockDim.x + threadIdx.x) >> 5;
  const int lane = threadIdx.x & 31;
  if (node >= N_NODES) return;
  const float dv = dinv[node];
  const int j0 = off[node], j1 = off[node + 1];

  float acc0 = h[(size_t)node * HID + lane] * dv * dv;  // self loop
  float acc1 = 0.0f, acc2 = 0.0f, acc3 = 0.0f;

  int j = j0;
  for (; j + 4 <= j1; j += 4) {
    const int s0 = csr_src[j + 0];
    const int s1 = csr_src[j + 1];
    const int s2 = csr_src[j + 2];
    const int s3 = csr_src[j + 3];
    const float w0 = dinv[s0] * dv;
    const float w1 = dinv[s1] * dv;
    const float w2 = dinv[s2] * dv;
    const float w3 = dinv[s3] * dv;
    acc0 += h[(size_t)s0 * HID + lane] * w0;
    acc1 += h[(size_t)s1 * HID + lane] * w1;
    acc2 += h[(size_t)s2 * HID + lane] * w2;
    acc3 += h[(size_t)s3 * HID + lane] * w3;
  }
  for (; j < j1; ++j) {
    const int s = csr_src[j];
    acc0 += h[(size_t)s * HID + lane] * (dinv[s] * dv);
  }

  const float r = (acc0 + acc1) + (acc2 + acc3) + bias[lane];
  out[(size_t)node * HID + lane] = r > 0.0f ? r : 0.0f;
}

// ----------------------------- fused MLP head (WMMA) -----------------------------
// out[m] = relu(a2[m]@Wl1 + bl1) @ Wl2 + bl2 ; one wave per 16 nodes

__global__ void __launch_bounds__(32) k_mlp_wmma(const float* __restrict__ A,
                                                 const float* __restrict__ Wl1,
                                                 const float* __restrict__ bl1,
                                                 const float* __restrict__ Wl2,
                                                 const float* __restrict__ bl2,
                                                 float* __restrict__ out) {
  const int m0   = blockIdx.x * 16;
  const int lane = threadIdx.x;
  const int half = lane >> 4;
  const int l    = lane & 15;
  v8f c0 = {};
  v8f c1 = {};
  const float* arow = A + (size_t)(m0 + l) * HID;
#pragma unroll
  for (int k = 0; k < HID; k += 4) {
    const int ka = k + 2 * half;
    v2f a;  a.x  = arow[ka];                a.y  = arow[ka + 1];
    v2f b0; b0.x = Wl1[ka * HID + l];       b0.y = Wl1[(ka + 1) * HID + l];
    v2f b1; b1.x = Wl1[ka * HID + 16 + l];  b1.y = Wl1[(ka + 1) * HID + 16 + l];
    c0 = __builtin_amdgcn_wmma_f32_16x16x4_f32(false, a, false, b0, (short)0, c0, false, false);
    c1 = __builtin_amdgcn_wmma_f32_16x16x4_f32(false, a, false, b1, (short)0, c1, false, false);
  }
  const float w2a = Wl2[l], w2b = Wl2[16 + l];
  const float bb0 = bl1[l], bb1 = bl1[16 + l];
  float p[8];
#pragma unroll
  for (int v = 0; v < 8; ++v) {
    float h0 = c0[v] + bb0; h0 = h0 > 0.0f ? h0 : 0.0f;
    float h1 = c1[v] + bb1; h1 = h1 > 0.0f ? h1 : 0.0f;
    p[v] = h0 * w2a + h1 * w2b;  // partial over this lane's two columns
  }
  // reduce the 16 lanes within each half (half 0 -> rows 0-7, half 1 -> rows 8-15)
#pragma unroll
  for (int m = 1; m < 16; m <<= 1) {
#pragma unroll
    for (int v = 0; v < 8; ++v) p[v] += __shfl_xor(p[v], m, 32);
  }
  if (l < 8) out[m0 + half * 8 + l] = p[l] + bl2[0];
}

// ----------------------------- launcher -----------------------------

extern "C" void kernel_launch(void* const* d_in, const int* in_sizes, int n_in,
                              void* d_out, int out_size, void* d_ws, size_t ws_size,
                              hipStream_t stream) {
  const float* x       = (const float*)d_in[0];
  const long long* ei  = (const long long*)d_in[1];   // int64 edge_index [2, E]
  const long long* src = ei;
  const long long* dst = ei + N_EDGES;
  const float* W1  = (const float*)d_in[2];
  const float* b1  = (const float*)d_in[3];
  const float* W2  = (const float*)d_in[4];
  const float* b2  = (const float*)d_in[5];
  const float* Wl1 = (const float*)d_in[6];
  const float* bl1 = (const float*)d_in[7];
  const float* Wl2 = (const float*)d_in[8];
  const float* bl2 = (const float*)d_in[9];
  float* out = (float*)d_out;

  // workspace carve-up (every buffer fully rewritten each call -> deterministic)
  char* ws = (char*)d_ws;
  auto carve = [&](size_t bytes) -> char* {
    char* p = ws;
    ws += (bytes + 255) & ~(size_t)255;
    return p;
  };
  float* bufA = (float*)carve(sizeof(float) * (size_t)N_NODES * HID);  // h1, then h2
  float* bufB = (float*)carve(sizeof(float) * (size_t)N_NODES * HID);  // a1, then a2
  float* dinv = (float*)carve(sizeof(float) * N_NODES);
  int* cnt    = (int*)carve(sizeof(int) * N_NODES);
  int* off    = (int*)carve(sizeof(int) * (N_NODES + 1));
  int* cur    = (int*)carve(sizeof(int) * N_NODES);
  int* bsum   = (int*)carve(sizeof(int) * NBLK);
  int* csr    = (int*)carve(sizeof(int) * (size_t)N_EDGES);
  (void)ws_size; (void)in_sizes; (void)n_in; (void)out_size;

  // --- CSR build (reused by both conv layers) ---
  k_zero_int<<<(N_NODES + 255) / 256, 256, 0, stream>>>(cnt, N_NODES);
  k_count<<<(N_EDGES + 255) / 256, 256, 0, stream>>>(dst, cnt);
  k_dinv<<<(N_NODES + 255) / 256, 256, 0, stream>>>(cnt, dinv);
  k_block_sums<<<NBLK, SCAN_B, 0, stream>>>(cnt, bsum);
  k_scan_sums<<<1, 1, 0, stream>>>(bsum);
  k_scan_final<<<NBLK, SCAN_B, 0, stream>>>(cnt, bsum, off, cur);
  k_csr_fill<<<(N_EDGES + 255) / 256, 256, 0, stream>>>(src, dst, cur, csr);

  // --- layer 1: h1 = X@W1 ; a1 = relu(agg(h1)+b1) ---
  k_gemm_wmma<IN_CH><<<N_NODES / 16, 32, 0, stream>>>(x, W1, bufA);
  k_aggregate<<<(N_NODES * 32) / 256, 256, 0, stream>>>(bufA, dinv, off, csr, b1, bufB);

  // --- layer 2: h2 = a1@W2 ; a2 = relu(agg(h2)+b2) ---
  k_gemm_wmma<HID><<<N_NODES / 16, 32, 0, stream>>>(bufB, W2, bufA);
  k_aggregate<<<(N_NODES * 32) / 256, 256, 0, stream>>>(bufA, dinv, off, csr, b2, bufB);

  // --- MLP head ---
  k_mlp_wmma<<<N_NODES / 16, 32, 0, stream>>>(bufB, Wl1, bl1, Wl2, bl2, out);
}